// Atten_8658654068993
// MI455X (gfx1250) — compile-verified
//
#include <hip/hip_runtime.h>
#include <hip/hip_bf16.h>

typedef __attribute__((ext_vector_type(2))) float v2f;
typedef __attribute__((ext_vector_type(8))) float v8f;
typedef __attribute__((ext_vector_type(4))) unsigned int u32x4;
typedef __attribute__((ext_vector_type(4))) int i32x4;
typedef __attribute__((ext_vector_type(8))) int i32x8;

#define Bb 256
#define Tt 64
#define Kk 49
#define Hh 512

__device__ __forceinline__ float fast_tanh(float x) {
#if __has_builtin(__builtin_amdgcn_tanhf)
    return __builtin_amdgcn_tanhf(x);
#else
    // tanh(x) = 1 - 2/(1+exp(2x)); exp via v_exp_f32, division via v_rcp_f32
    float e = __builtin_amdgcn_exp2f(x * 2.885390081777927f);   // exp2(2x*log2 e)
    return 1.0f - 2.0f * __builtin_amdgcn_rcpf(e + 1.0f);
#endif
}

__device__ __forceinline__ float fast_exp(float x) {
    return __builtin_amdgcn_exp2f(x * 1.4426950408889634f);
}

// ---------------------------------------------------------------------------
// Tensor Data Mover: 2D tile load (tile_cols out of row_stride-wide rows,
// nrows rows) from global into contiguous LDS rows. D# per ISA 08 §8.
// ---------------------------------------------------------------------------
#if __has_builtin(__builtin_amdgcn_tensor_load_to_lds)
#define HAVE_TDM 1
__device__ __forceinline__ void tdm_load_2d(unsigned lds_off, const float* gptr,
                                            int nrows, int tile_cols, int row_stride) {
    unsigned long long ga = (unsigned long long)(uintptr_t)gptr;
    u32x4 g0;
    g0[0] = 1u;                                            // count=1 (valid user D#)
    g0[1] = lds_off;                                       // LDS byte address
    g0[2] = (unsigned)(ga & 0xFFFFFFFFu);                  // global addr [31:0]
    g0[3] = (unsigned)((ga >> 32) & 0x1FFFFFFu)            // global addr [56:32]
            | (2u << 30);                                  // type = 2 ("image")
    i32x8 g1;
    g1[0] = 0x20000;                                       // data_size = 2 (4 bytes)
    g1[1] = (row_stride & 0xFFFF) << 16;                   // tensor_dim0 [15:0]
    g1[2] = ((row_stride >> 16) & 0xFFFF)                  // tensor_dim0 [31:16]
            | ((nrows & 0xFFFF) << 16);                    // tensor_dim1 [15:0]
    g1[3] = ((nrows >> 16) & 0xFFFF)                       // tensor_dim1 [31:16]
            | ((tile_cols & 0xFFFF) << 16);                // tile_dim0
    g1[4] = nrows & 0xFFFF;                                // tile_dim1 (tile_dim2=0)
    g1[5] = row_stride;                                    // tensor_dim0_stride [31:0]
    g1[6] = 0;                                             // stride hi / dim1_stride
    g1[7] = 0;
    i32x4 z4 = {0, 0, 0, 0};
#if __clang_major__ >= 23
    i32x8 z8 = {0, 0, 0, 0, 0, 0, 0, 0};
    __builtin_amdgcn_tensor_load_to_lds(g0, g1, z4, z4, z8, 0);
#else
    __builtin_amdgcn_tensor_load_to_lds(g0, g1, z4, z4, 0);
#endif
}
__device__ __forceinline__ void tdm_wait() {
#if __has_builtin(__builtin_amdgcn_s_wait_tensorcnt)
    __builtin_amdgcn_s_wait_tensorcnt(0);
#else
    asm volatile("s_wait_tensorcnt 0x0" ::: "memory");
#endif
}
#endif

// ---------------------------------------------------------------------------
// Kernel 1: C[M x 64] = X[M x 512] * W[49 x 512]^T   (cols 49..63 forced to 0)
// One wave computes one 16x16 tile across 4 N-tiles, V_WMMA_F32_16X16X4_F32.
// ---------------------------------------------------------------------------
__global__ __launch_bounds__(256) void proj_gemm_wmma(
    const float* __restrict__ X, const float* __restrict__ W,
    float* __restrict__ out, int Mtiles)
{
    const int wave = threadIdx.x >> 5;
    const int lane = threadIdx.x & 31;
    const int mt = blockIdx.x * 8 + wave;
    if (mt >= Mtiles) return;                      // wave-uniform

    const int lr    = lane & 15;
    const int khalf = (lane >> 4) * 2;             // 0 or 2

    const float* xrow = X + (size_t)(mt * 16 + lr) * Hh + khalf;

    const float* wptr[4];
    float wmask[4];
#pragma unroll
    for (int nt = 0; nt < 4; ++nt) {
        int col = nt * 16 + lr;
        wmask[nt] = (col < Kk) ? 1.0f : 0.0f;
        int ccol  = (col < Kk) ? col : (Kk - 1);
        wptr[nt]  = W + (size_t)ccol * Hh + khalf;
    }

    v8f acc[4] = {};
    for (int k0 = 0; k0 < Hh; k0 += 4) {
        v2f a = *(const v2f*)(xrow + k0);
#pragma unroll
        for (int nt = 0; nt < 4; ++nt) {
            v2f bf = *(const v2f*)(wptr[nt] + k0);
            bf *= wmask[nt];
            acc[nt] = __builtin_amdgcn_wmma_f32_16x16x4_f32(
                false, a, false, bf, (short)0, acc[nt], false, false);
        }
    }

    const int mbase = mt * 16 + ((lane >> 4) ? 8 : 0);
#pragma unroll
    for (int r = 0; r < 8; ++r) {
        float* orow = out + (size_t)(mbase + r) * 64;
#pragma unroll
        for (int nt = 0; nt < 4; ++nt)
            orow[nt * 16 + lr] = acc[nt][r];
    }
}

// ---------------------------------------------------------------------------
// Kernel 2: per-batch attention core. 256 threads (8 waves) per block.
// ---------------------------------------------------------------------------
__global__ __launch_bounds__(256) void attn_core(
    const float* __restrict__ V, const float* __restrict__ s_t,
    const float* __restrict__ cg, const float* __restrict__ cs,
    const float* __restrict__ cv, const float* __restrict__ Wh,
    float* __restrict__ out_chat, float* __restrict__ out_alpha,
    float* __restrict__ out_beta)
{
    __shared__ __align__(16) float sCG[Tt * Kk];       // 12.25 KB (compacted rows)
    __shared__ __align__(16) float sCV[Kk * Kk];       // 9.4 KB
    __shared__ __align__(16) float sZ[Tt * 52];        // 13 KB (alpha, K padded)
    __shared__ __align__(16) float sWh[Kk];
    __shared__ float sBeta[Tt];

    const int b   = blockIdx.x;
    const int tid = threadIdx.x;
    const float* Vb = V + (size_t)b * Kk * Hh;

    // ---- phase 0: stage per-batch tensors into LDS ----------------------
#ifdef HAVE_TDM
    if (tid < 32) {                                   // wave 0 only: one TDM issue each
        tdm_load_2d((unsigned)(uintptr_t)(void*)sCG,
                    cg + (size_t)b * (Tt * 64), Tt, Kk, 64);
        tdm_load_2d((unsigned)(uintptr_t)(void*)sCV,
                    cv + (size_t)b * (Kk * 64), Kk, Kk, 64);
        tdm_wait();
    }
#else
    for (int i = tid; i < Tt * Kk; i += 256) {
        int t = i / Kk, a = i % Kk;
        sCG[i] = cg[(size_t)b * (Tt * 64) + t * 64 + a];
    }
    for (int i = tid; i < Kk * Kk; i += 256) {
        int k = i / Kk, a = i % Kk;
        sCV[i] = cv[(size_t)b * (Kk * 64) + k * 64 + a];
    }
#endif
    if (tid < Kk) sWh[tid] = Wh[tid];
    if (tid < Tt * 3) {                               // zero K-padding of alpha
        int t = tid / 3, p = tid % 3;
        sZ[t * 52 + Kk + p] = 0.0f;
    }
    __syncthreads();

    // ---- phase 1: z[t][k] = sum_a Wh[a]*tanh(cg[t,a]+cv[k,a]) -----------
    for (int e = tid; e < Tt * Kk; e += 256) {
        int t = e / Kk, k = e % Kk;
        const float* cgp = &sCG[t * Kk];
        const float* cvp = &sCV[k * Kk];
        float acc = 0.0f;
#pragma unroll 7
        for (int a = 0; a < Kk; ++a)
            acc += sWh[a] * fast_tanh(cgp[a] + cvp[a]);
        sZ[t * 52 + k] = acc;
    }
    __syncthreads();

    // ---- phase 2: softmaxes --------------------------------------------
    if (tid < Tt) {
        const int t = tid;
        float* zp = &sZ[t * 52];
        float m = -3.4e38f;
        for (int k = 0; k < Kk; ++k) m = fmaxf(m, zp[k]);
        float s = 0.0f;
        for (int k = 0; k < Kk; ++k) { float e = fast_exp(zp[k] - m); zp[k] = e; s += e; }
        const float inv = 1.0f / s;
        float* oA = out_alpha + ((size_t)b * Tt + t) * Kk;
        for (int k = 0; k < Kk; ++k) { float al = zp[k] * inv; zp[k] = al; oA[k] = al; }

        // z_ext = sum_a Wh[a]*tanh(cs_raw + cg)
        const float* csp = cs + (size_t)b * (Tt * 64) + t * 64;
        const float* cgp = &sCG[t * Kk];
        float ze = 0.0f;
        for (int a = 0; a < Kk; ++a) ze += sWh[a] * fast_tanh(csp[a] + cgp[a]);

        float m2 = fmaxf(m, ze);
        float ee = fast_exp(ze - m2);
        float beta = ee / (s * fast_exp(m - m2) + ee);
        sBeta[t] = beta;
        out_beta[(size_t)b * Tt + t] = beta;
    }
    __syncthreads();

    // ---- phase 3: c_t = alpha @ V_b (WMMA f32 16x16x4) + beta blend -----
    const int wave = tid >> 5;
    const int lane = tid & 31;
    const int lr = lane & 15;
    const int lh = lane >> 4;                          // 0 or 1

    for (int tile = wave; tile < 4 * 32; tile += 8) {  // uniform across wave
        const int t0 = (tile >> 5) * 16;               // 4 t-tiles
        const int h0 = (tile & 31) * 16;               // 32 h-tiles
        const int h  = h0 + lr;

        v8f acc = {};
        for (int k0 = 0; k0 < 52; k0 += 4) {
            const int ka = k0 + 2 * lh;
            v2f a = *(const v2f*)(&sZ[(t0 + lr) * 52 + ka]);
            const int r0 = (ka     < Kk) ? ka     : (Kk - 1);
            const int r1 = (ka + 1 < Kk) ? ka + 1 : (Kk - 1);
            v2f bv;
            bv.x = Vb[(size_t)r0 * Hh + h] * ((ka     < Kk) ? 1.0f : 0.0f);
            bv.y = Vb[(size_t)r1 * Hh + h] * ((ka + 1 < Kk) ? 1.0f : 0.0f);
            acc = __builtin_amdgcn_wmma_f32_16x16x4_f32(
                false, a, false, bv, (short)0, acc, false, false);
        }

#pragma unroll
        for (int r = 0; r < 8; ++r) {
            const int t = t0 + r + 8 * lh;
            const float bt = sBeta[t];
            const size_t off = ((size_t)b * Tt + t) * Hh + h;
            out_chat[off] = bt * s_t[off] + (1.0f - bt) * acc[r];
        }
    }
}

// ---------------------------------------------------------------------------
extern "C" void kernel_launch(void* const* d_in, const int* in_sizes, int n_in,
                              void* d_out, int out_size, void* d_ws, size_t ws_size,
                              hipStream_t stream) {
    const float* V   = (const float*)d_in[0];
    const float* h_t = (const float*)d_in[1];
    const float* s_t = (const float*)d_in[2];
    const float* Wv  = (const float*)d_in[3];
    const float* Wg  = (const float*)d_in[4];
    const float* Ws  = (const float*)d_in[5];
    const float* Wh  = (const float*)d_in[6];

    float* out       = (float*)d_out;
    float* out_chat  = out;                                   // B*T*H
    float* out_alpha = out + (size_t)Bb * Tt * Hh;            // B*T*K
    float* out_beta  = out_alpha + (size_t)Bb * Tt * Kk;      // B*T

    float* ws = (float*)d_ws;
    float* cg = ws;                                           // B*64*64
    float* cs = cg + (size_t)Bb * Tt * 64;                    // B*64*64
    float* cv = cs + (size_t)Bb * Tt * 64;                    // B*49*64

    proj_gemm_wmma<<<128, 256, 0, stream>>>(h_t, Wg, cg, 1024);
    proj_gemm_wmma<<<128, 256, 0, stream>>>(s_t, Ws, cs, 1024);
    proj_gemm_wmma<<< 98, 256, 0, stream>>>(V,   Wv, cv,  784);

    attn_core<<<Bb, 256, 0, stream>>>(V, s_t, cg, cs, cv, Wh,
                                      out_chat, out_alpha, out_beta);
}